// PatchedLlamaAttention_57243324121573
// MI455X (gfx1250) — compile-verified
//
#include <hip/hip_runtime.h>
#include <hip/hip_bf16.h>

// ---------------------------------------------------------------------------
// PatchedLlamaAttention for MI455X (gfx1250): fp32 WMMA (v_wmma_f32_16x16x4_f32)
// B=1, S=2048, DM=2048, H=16, HKV=4, HD=128, alpha-entmax(1.5) => p=max(X-tau,0)^2
// GEMMs software-pipelined (register double-buffered fragments, 8-deep K,
// unroll-2 => distance-2 prefetch, no rotation moves).
// ---------------------------------------------------------------------------

typedef float v2f __attribute__((ext_vector_type(2)));
typedef float v8f __attribute__((ext_vector_type(8)));

#define S_   2048
#define DM_  2048
#define H_   16
#define HKV_ 4
#define HD_  128
#define SCALING_ 0.08838834764831845f   // 128^-0.5

// D = A(16x4) * B(4x16) + C, all f32, wave32.
__device__ __forceinline__ v8f wmma4(v2f a, v2f b, v8f c) {
  return __builtin_amdgcn_wmma_f32_16x16x4_f32(
      /*neg_a=*/false, a, /*neg_b=*/false, b,
      /*c_mod=*/(short)0, c, /*reuse_a=*/false, /*reuse_b=*/false);
}

// ---------------------------------------------------------------------------
// C[M,N] = A[M,Kd] * W[N,Kd]^T, 16-row wave tile.
// block = 128 threads (4 waves); block tile 64x64; grid = (N/64, M/64).
// ---------------------------------------------------------------------------
__global__ void __launch_bounds__(128) gemm_nt_kernel(
    const float* __restrict__ A, const float* __restrict__ W,
    float* __restrict__ C, int Kd, int lda, int ldw, int ldc) {
  const int lane = threadIdx.x & 31;
  const int wave = threadIdx.x >> 5;
  const int lm   = lane & 15;
  const int kh   = lane >> 4;
  const int m0   = blockIdx.y * 64 + wave * 16;
  const int n0   = blockIdx.x * 64;

  const float* Arow = A + (size_t)(m0 + lm) * lda + 2 * kh;
  const float* Wr[4];
#pragma unroll
  for (int j = 0; j < 4; ++j)
    Wr[j] = W + (size_t)(n0 + 16 * j + lm) * ldw + 2 * kh;

  v8f acc[4] = {};
  v2f a0 = *(const v2f*)(Arow + 0);
  v2f a1 = *(const v2f*)(Arow + 4);
  v2f b0[4], b1[4];
#pragma unroll
  for (int j = 0; j < 4; ++j) {
    b0[j] = *(const v2f*)(Wr[j] + 0);
    b1[j] = *(const v2f*)(Wr[j] + 4);
  }

  int k = 0;
#pragma unroll 2
  for (; k + 16 <= Kd; k += 8) {
    v2f na0 = *(const v2f*)(Arow + k + 8);
    v2f na1 = *(const v2f*)(Arow + k + 12);
    v2f nb0[4], nb1[4];
#pragma unroll
    for (int j = 0; j < 4; ++j) {
      nb0[j] = *(const v2f*)(Wr[j] + k + 8);
      nb1[j] = *(const v2f*)(Wr[j] + k + 12);
    }
#pragma unroll
    for (int j = 0; j < 4; ++j) acc[j] = wmma4(a0, b0[j], acc[j]);
#pragma unroll
    for (int j = 0; j < 4; ++j) acc[j] = wmma4(a1, b1[j], acc[j]);
    a0 = na0; a1 = na1;
#pragma unroll
    for (int j = 0; j < 4; ++j) { b0[j] = nb0[j]; b1[j] = nb1[j]; }
  }
#pragma unroll
  for (int j = 0; j < 4; ++j) acc[j] = wmma4(a0, b0[j], acc[j]);
#pragma unroll
  for (int j = 0; j < 4; ++j) acc[j] = wmma4(a1, b1[j], acc[j]);

#pragma unroll
  for (int j = 0; j < 4; ++j)
#pragma unroll
    for (int i = 0; i < 8; ++i)
      C[(size_t)(m0 + 8 * kh + i) * ldc + n0 + 16 * j + lm] = acc[j][i];
}

// ---------------------------------------------------------------------------
// C[M,N] = A[M,Kd] * W[N,Kd]^T, 32-row wave tile (higher arithmetic intensity).
// block = 128 threads (4 waves); block tile 128x64; grid = (N/64, M/128).
// ---------------------------------------------------------------------------
__global__ void __launch_bounds__(128) gemm_nt32_kernel(
    const float* __restrict__ A, const float* __restrict__ W,
    float* __restrict__ C, int Kd, int lda, int ldw, int ldc) {
  const int lane = threadIdx.x & 31;
  const int wave = threadIdx.x >> 5;
  const int lm   = lane & 15;
  const int kh   = lane >> 4;
  const int m0   = blockIdx.y * 128 + wave * 32;
  const int n0   = blockIdx.x * 64;

  const float* Ar0 = A + (size_t)(m0 + lm) * lda + 2 * kh;
  const float* Ar1 = Ar0 + (size_t)16 * lda;
  const float* Wr[4];
#pragma unroll
  for (int j = 0; j < 4; ++j)
    Wr[j] = W + (size_t)(n0 + 16 * j + lm) * ldw + 2 * kh;

  v8f acc0[4] = {};
  v8f acc1[4] = {};
  // a{row-group}{k-substep}
  v2f a00 = *(const v2f*)(Ar0 + 0);
  v2f a01 = *(const v2f*)(Ar0 + 4);
  v2f a10 = *(const v2f*)(Ar1 + 0);
  v2f a11 = *(const v2f*)(Ar1 + 4);
  v2f b0[4], b1[4];
#pragma unroll
  for (int j = 0; j < 4; ++j) {
    b0[j] = *(const v2f*)(Wr[j] + 0);
    b1[j] = *(const v2f*)(Wr[j] + 4);
  }

  int k = 0;
#pragma unroll 2
  for (; k + 16 <= Kd; k += 8) {
    v2f na00 = *(const v2f*)(Ar0 + k + 8);
    v2f na01 = *(const v2f*)(Ar0 + k + 12);
    v2f na10 = *(const v2f*)(Ar1 + k + 8);
    v2f na11 = *(const v2f*)(Ar1 + k + 12);
    v2f nb0[4], nb1[4];
#pragma unroll
    for (int j = 0; j < 4; ++j) {
      nb0[j] = *(const v2f*)(Wr[j] + k + 8);
      nb1[j] = *(const v2f*)(Wr[j] + k + 12);
    }
#pragma unroll
    for (int j = 0; j < 4; ++j) {
      acc0[j] = wmma4(a00, b0[j], acc0[j]);
      acc1[j] = wmma4(a10, b0[j], acc1[j]);
    }
#pragma unroll
    for (int j = 0; j < 4; ++j) {
      acc0[j] = wmma4(a01, b1[j], acc0[j]);
      acc1[j] = wmma4(a11, b1[j], acc1[j]);
    }
    a00 = na00; a01 = na01; a10 = na10; a11 = na11;
#pragma unroll
    for (int j = 0; j < 4; ++j) { b0[j] = nb0[j]; b1[j] = nb1[j]; }
  }
#pragma unroll
  for (int j = 0; j < 4; ++j) {
    acc0[j] = wmma4(a00, b0[j], acc0[j]);
    acc1[j] = wmma4(a10, b0[j], acc1[j]);
  }
#pragma unroll
  for (int j = 0; j < 4; ++j) {
    acc0[j] = wmma4(a01, b1[j], acc0[j]);
    acc1[j] = wmma4(a11, b1[j], acc1[j]);
  }

#pragma unroll
  for (int j = 0; j < 4; ++j)
#pragma unroll
    for (int i = 0; i < 8; ++i) {
      C[(size_t)(m0 + 8 * kh + i) * ldc + n0 + 16 * j + lm]      = acc0[j][i];
      C[(size_t)(m0 + 16 + 8 * kh + i) * ldc + n0 + 16 * j + lm] = acc1[j][i];
    }
}

// ---------------------------------------------------------------------------
// RoPE in place on X[S][nh*128]; thread handles pair (d, d+64) of one head.
// ---------------------------------------------------------------------------
__global__ void __launch_bounds__(256) rope_kernel(
    float* __restrict__ X, const float* __restrict__ cosT,
    const float* __restrict__ sinT, int nh, int ld) {
  int idx = blockIdx.x * blockDim.x + threadIdx.x;
  int per_s = nh * 64;
  int s = idx / per_s;
  int r = idx - s * per_s;
  int h = r >> 6;
  int d = r & 63;
  float* p = X + (size_t)s * ld + h * HD_;
  float x1 = p[d];
  float x2 = p[d + 64];
  float c1 = cosT[s * HD_ + d],      s1 = sinT[s * HD_ + d];
  float c2 = cosT[s * HD_ + d + 64], s2 = sinT[s * HD_ + d + 64];
  p[d]      = x1 * c1 - x2 * s1;
  p[d + 64] = x2 * c2 + x1 * s2;
}

// ---------------------------------------------------------------------------
// scores[h][q][k] = scaling * (Q_h q . K_kv k), causal mask (-inf above diag)
// grid = (S/64, S/64, H); upper-triangular blocks skip the GEMM entirely.
// ---------------------------------------------------------------------------
__global__ void __launch_bounds__(128) scores_kernel(
    const float* __restrict__ Q, const float* __restrict__ K,
    float* __restrict__ attn) {
  const int h  = blockIdx.z;
  const int kv = h >> 2;                 // GQA group
  const int lane = threadIdx.x & 31;
  const int wave = threadIdx.x >> 5;
  const int lm = lane & 15;
  const int kh = lane >> 4;
  const int m0 = blockIdx.y * 64 + wave * 16;   // query rows
  const int n0 = blockIdx.x * 64;               // key cols
  float* out = attn + (size_t)h * S_ * S_;

  if (blockIdx.x > blockIdx.y) {          // fully masked tile
#pragma unroll
    for (int j = 0; j < 4; ++j)
#pragma unroll
      for (int i = 0; i < 8; ++i)
        out[(size_t)(m0 + 8 * kh + i) * S_ + n0 + 16 * j + lm] =
            -__builtin_inff();
    return;
  }

  const float* Arow = Q + (size_t)(m0 + lm) * DM_ + h * HD_ + 2 * kh;
  const float* Kr[4];
#pragma unroll
  for (int j = 0; j < 4; ++j)
    Kr[j] = K + (size_t)(n0 + 16 * j + lm) * (HKV_ * HD_) + kv * HD_ + 2 * kh;

  v8f acc[4] = {};
  v2f a0 = *(const v2f*)(Arow + 0);
  v2f a1 = *(const v2f*)(Arow + 4);
  v2f b0[4], b1[4];
#pragma unroll
  for (int j = 0; j < 4; ++j) {
    b0[j] = *(const v2f*)(Kr[j] + 0);
    b1[j] = *(const v2f*)(Kr[j] + 4);
  }
  int k = 0;
#pragma unroll 2
  for (; k + 16 <= HD_; k += 8) {
    v2f na0 = *(const v2f*)(Arow + k + 8);
    v2f na1 = *(const v2f*)(Arow + k + 12);
    v2f nb0[4], nb1[4];
#pragma unroll
    for (int j = 0; j < 4; ++j) {
      nb0[j] = *(const v2f*)(Kr[j] + k + 8);
      nb1[j] = *(const v2f*)(Kr[j] + k + 12);
    }
#pragma unroll
    for (int j = 0; j < 4; ++j) acc[j] = wmma4(a0, b0[j], acc[j]);
#pragma unroll
    for (int j = 0; j < 4; ++j) acc[j] = wmma4(a1, b1[j], acc[j]);
    a0 = na0; a1 = na1;
#pragma unroll
    for (int j = 0; j < 4; ++j) { b0[j] = nb0[j]; b1[j] = nb1[j]; }
  }
#pragma unroll
  for (int j = 0; j < 4; ++j) acc[j] = wmma4(a0, b0[j], acc[j]);
#pragma unroll
  for (int j = 0; j < 4; ++j) acc[j] = wmma4(a1, b1[j], acc[j]);

#pragma unroll
  for (int j = 0; j < 4; ++j)
#pragma unroll
    for (int i = 0; i < 8; ++i) {
      int row = m0 + 8 * kh + i;
      int col = n0 + 16 * j + lm;
      float v = acc[j][i] * SCALING_;
      out[(size_t)row * S_ + col] = (col > row) ? -__builtin_inff() : v;
    }
}

// ---------------------------------------------------------------------------
// alpha-entmax (alpha=1.5) bisection, in place. One WAVE per row of 2048:
// 64 elements/lane live in VGPRs for all 50 iterations; shfl-only reductions,
// zero barriers. Block = 256 threads = 8 independent rows.
// ---------------------------------------------------------------------------
__device__ __forceinline__ float wave_sum_b(float v) {
#pragma unroll
  for (int off = 16; off > 0; off >>= 1) v += __shfl_down(v, off, 32);
  return __shfl(v, 0, 32);
}
__device__ __forceinline__ float wave_max_b(float v) {
#pragma unroll
  for (int off = 16; off > 0; off >>= 1) v = fmaxf(v, __shfl_down(v, off, 32));
  return __shfl(v, 0, 32);
}

__global__ void __launch_bounds__(256) entmax_kernel(float* __restrict__ attn) {
  const int lane = threadIdx.x & 31;
  const int wid  = threadIdx.x >> 5;
  float* row = attn + ((size_t)blockIdx.x * 8 + wid) * S_;

  float v[64];
#pragma unroll
  for (int i = 0; i < 64; ++i) v[i] = row[lane + 32 * i] * 0.5f;  // X = x*(a-1)

  float m = v[0];
#pragma unroll
  for (int i = 1; i < 64; ++i) m = fmaxf(m, v[i]);
  const float rowmax = wave_max_b(m);

  float tau_lo = rowmax - 1.0f;
  float dm = 0.9779029130912362f;   // 1 - (1/2048)^0.5
  float tau_m = tau_lo;

  for (int it = 0; it < 50; ++it) {
    dm *= 0.5f;
    tau_m = tau_lo + dm;
    float s = 0.0f;
#pragma unroll
    for (int i = 0; i < 64; ++i) {
      float t = fmaxf(v[i] - tau_m, 0.0f);
      s = fmaf(t, t, s);
    }
    s = wave_sum_b(s);
    if (s - 1.0f >= 0.0f) tau_lo = tau_m;
  }

  // final p with last tau_m, then normalize (overwrite v with p)
  float s = 0.0f;
#pragma unroll
  for (int i = 0; i < 64; ++i) {
    float t = fmaxf(v[i] - tau_m, 0.0f);
    v[i] = t * t;
    s += v[i];
  }
  s = wave_sum_b(s);
  const float inv = 1.0f / s;
#pragma unroll
  for (int i = 0; i < 64; ++i) row[lane + 32 * i] = v[i] * inv;
}

// ---------------------------------------------------------------------------
// OH[q][h*128+d] = sum_k attn[h][q][k] * V[k][kv*128+d]   ("NN" GEMM per head)
// grid = (HD/64=2, S/64=32, H). Causal: attn[q][k]==0 for k>q, so the K loop
// stops at m0+16 (wave-uniform => EXEC stays all-ones for WMMA).
// ---------------------------------------------------------------------------
__global__ void __launch_bounds__(128) av_kernel(
    const float* __restrict__ attn, const float* __restrict__ V,
    float* __restrict__ OH) {
  const int h  = blockIdx.z;
  const int kv = h >> 2;
  const int lane = threadIdx.x & 31;
  const int wave = threadIdx.x >> 5;
  const int lm = lane & 15;
  const int kh = lane >> 4;
  const int m0 = blockIdx.y * 64 + wave * 16;
  const int n0 = blockIdx.x * 64;
  const int ldb = HKV_ * HD_;            // 512

  const float* Arow = attn + (size_t)h * S_ * S_ + (size_t)(m0 + lm) * S_ + 2 * kh;
  const float* Bc   = V + kv * HD_ + n0 + lm;   // column base, index [k][.]

  const int kmax = m0 + 16;              // multiple of 16, >= 16

  v8f acc[4] = {};
  v2f a0 = *(const v2f*)(Arow + 0);
  v2f a1 = *(const v2f*)(Arow + 4);
  v2f b0[4], b1[4];
#pragma unroll
  for (int j = 0; j < 4; ++j) {
    const float* B0 = Bc + (size_t)(0 + 2 * kh) * ldb + 16 * j;
    const float* B1 = Bc + (size_t)(4 + 2 * kh) * ldb + 16 * j;
    b0[j].x = B0[0];  b0[j].y = B0[ldb];
    b1[j].x = B1[0];  b1[j].y = B1[ldb];
  }

  int k = 0;
#pragma unroll 2
  for (; k + 16 <= kmax; k += 8) {
    v2f na0 = *(const v2f*)(Arow + k + 8);
    v2f na1 = *(const v2f*)(Arow + k + 12);
    v2f nb0[4], nb1[4];
#pragma unroll
    for (int j = 0; j < 4; ++j) {
      const float* B0 = Bc + (size_t)(k + 8 + 2 * kh) * ldb + 16 * j;
      const float* B1 = Bc + (size_t)(k + 12 + 2 * kh) * ldb + 16 * j;
      nb0[j].x = B0[0];  nb0[j].y = B0[ldb];
      nb1[j].x = B1[0];  nb1[j].y = B1[ldb];
    }
#pragma unroll
    for (int j = 0; j < 4; ++j) acc[j] = wmma4(a0, b0[j], acc[j]);
#pragma unroll
    for (int j = 0; j < 4; ++j) acc[j] = wmma4(a1, b1[j], acc[j]);
    a0 = na0; a1 = na1;
#pragma unroll
    for (int j = 0; j < 4; ++j) { b0[j] = nb0[j]; b1[j] = nb1[j]; }
  }
#pragma unroll
  for (int j = 0; j < 4; ++j) acc[j] = wmma4(a0, b0[j], acc[j]);
#pragma unroll
  for (int j = 0; j < 4; ++j) acc[j] = wmma4(a1, b1[j], acc[j]);

#pragma unroll
  for (int j = 0; j < 4; ++j)
#pragma unroll
    for (int i = 0; i < 8; ++i)
      OH[(size_t)(m0 + 8 * kh + i) * (H_ * HD_) + h * HD_ + n0 + 16 * j + lm] =
          acc[j][i];
}

// ---------------------------------------------------------------------------
// launch
// ---------------------------------------------------------------------------
extern "C" void kernel_launch(void* const* d_in, const int* in_sizes, int n_in,
                              void* d_out, int out_size, void* d_ws,
                              size_t ws_size, hipStream_t stream) {
  const float* hs   = (const float*)d_in[0];  // [S, DM]
  const float* cosT = (const float*)d_in[1];  // [1, S, HD]
  const float* sinT = (const float*)d_in[2];
  const float* wq   = (const float*)d_in[3];  // [H*HD, DM]
  const float* wk   = (const float*)d_in[4];  // [HKV*HD, DM]
  const float* wv   = (const float*)d_in[5];
  const float* wo   = (const float*)d_in[6];  // [DM, H*HD]

  float* out  = (float*)d_out;                           // [S, DM]
  float* attn = (float*)d_out + (size_t)S_ * DM_;        // [H, S, S]

  float* ws = (float*)d_ws;                    // 40 MB used
  float* Q  = ws;                              // [S, H*HD]   4M floats
  float* K  = Q + (size_t)S_ * (H_ * HD_);     // [S, HKV*HD] 1M floats
  float* V  = K + (size_t)S_ * (HKV_ * HD_);   // [S, HKV*HD] 1M floats
  float* OH = V + (size_t)S_ * (HKV_ * HD_);   // [S, H*HD]   4M floats

  dim3 blk(128);

  // 1) projections: X @ W^T   (Q: 32-row tile; K/V: 16-row tile for occupancy)
  gemm_nt32_kernel<<<dim3((H_ * HD_) / 64, S_ / 128), blk, 0, stream>>>(
      hs, wq, Q, DM_, DM_, DM_, H_ * HD_);
  gemm_nt_kernel<<<dim3((HKV_ * HD_) / 64, S_ / 64), blk, 0, stream>>>(
      hs, wk, K, DM_, DM_, DM_, HKV_ * HD_);
  gemm_nt_kernel<<<dim3((HKV_ * HD_) / 64, S_ / 64), blk, 0, stream>>>(
      hs, wv, V, DM_, DM_, DM_, HKV_ * HD_);

  // 2) RoPE in place on Q and K
  rope_kernel<<<(S_ * H_ * 64) / 256, 256, 0, stream>>>(Q, cosT, sinT, H_,
                                                        H_ * HD_);
  rope_kernel<<<(S_ * HKV_ * 64) / 256, 256, 0, stream>>>(K, cosT, sinT, HKV_,
                                                          HKV_ * HD_);

  // 3) scaled causal scores -> attn region of d_out
  scores_kernel<<<dim3(S_ / 64, S_ / 64, H_), blk, 0, stream>>>(Q, K, attn);

  // 4) entmax bisection in place (one wave per attention row)
  entmax_kernel<<<(H_ * S_) / 8, 256, 0, stream>>>(attn);

  // 5) attn @ V -> OH in [S][H*HD] layout (causal K-limit)
  av_kernel<<<dim3(HD_ / 64, S_ / 64, H_), blk, 0, stream>>>(attn, V, OH);

  // 6) output projection: OH @ wo^T -> out  (32-row tile)
  gemm_nt32_kernel<<<dim3(DM_ / 64, S_ / 128), blk, 0, stream>>>(
      OH, wo, out, H_ * HD_, H_ * HD_, H_ * HD_, DM_);
}